// plot_calibration_45818711114206
// MI455X (gfx1250) — compile-verified
//
#include <hip/hip_runtime.h>
#include <stdint.h>

#define NBINS     10
#define C         10
#define TPB       256
#define TILE_ROWS 256
#define NBLK_MAX  1024

// fp32 bin boundaries = linspace(0,1,11) rounded to fp32
__device__ __constant__ float kBound[NBINS + 1] = {
    0.0f, 0.1f, 0.2f, 0.3f, 0.4f, 0.5f, 0.6f, 0.7f, 0.8f, 0.9f, 1.0f};

// ---- CDNA5 async global->LDS copies (ASYNCcnt-tracked, CDNA5 ISA) ----
__device__ __forceinline__ void async_ld_b128(unsigned lds_off, const void* gaddr) {
    asm volatile("global_load_async_to_lds_b128 %0, %1, off"
                 :: "v"(lds_off), "v"(gaddr) : "memory");
}
__device__ __forceinline__ void async_ld_b64(unsigned lds_off, const void* gaddr) {
    asm volatile("global_load_async_to_lds_b64 %0, %1, off"
                 :: "v"(lds_off), "v"(gaddr) : "memory");
}
__device__ __forceinline__ void wait_async_le(int) = delete;
__device__ __forceinline__ void wait_async0() {
    asm volatile("s_wait_asynccnt 0x0" ::: "memory");
}
__device__ __forceinline__ void wait_async3() {   // oldest tile done, next tile in flight
    asm volatile("s_wait_asynccnt 0x3" ::: "memory");
}
__device__ __forceinline__ unsigned ballot32(bool p) {
#if __has_builtin(__builtin_amdgcn_ballot_w32)
    return __builtin_amdgcn_ballot_w32(p);
#else
    return (unsigned)__ballot(p);
#endif
}

// Per-row classification: returns bin in [0,10) or -1 (invalid), sets `good`.
__device__ __forceinline__ int classify_row(const float* __restrict__ row,
                                            int label, bool& good) {
    float bv = row[0];
    int best = 0;
#pragma unroll
    for (int j = 1; j < C; ++j) {
        float v = row[j];
        if (v > bv) { bv = v; best = j; }   // first-max argmax (strict >)
    }
    float conf = bv - 1e-7f;                // sm = logits - EPS
    int cnt_lt = 0;
#pragma unroll
    for (int j = 0; j <= NBINS; ++j) cnt_lt += (kBound[j] < conf) ? 1 : 0;
    int bin = cnt_lt - 1;                   // searchsorted(left) - 1
    bool valid = ((unsigned)bin < (unsigned)NBINS);
    good = valid && (best == label);
    return valid ? bin : -1;
}

__global__ void __launch_bounds__(TPB) calib_main(const float* __restrict__ logits,
                                                  const int* __restrict__ labels,
                                                  unsigned* __restrict__ partials,
                                                  int n) {
    __shared__ __align__(16) float tile[2][TILE_ROWS * C];  // 2 x 10240 B
    __shared__ unsigned s_cnt[NBINS];
    __shared__ unsigned s_cor[NBINS];

    const int tid  = threadIdx.x;
    const int lane = tid & 31;
    if (tid < NBINS) { s_cnt[tid] = 0u; s_cor[tid] = 0u; }
    __syncthreads();

    const int fullTiles = n / TILE_ROWS;
    const int g = gridDim.x;

    // Issue 3 coalesced async ops (2x b128 + 1x b64 = 2560 dwords) for tile t into buf.
    auto issue_tile = [&](const float* buf, int t) {
        const unsigned ldsB = (unsigned)(uintptr_t)buf;
        const size_t base = (size_t)t * (TILE_ROWS * C);   // dword index
        async_ld_b128(ldsB + (unsigned)tid * 16u,
                      logits + base + (size_t)tid * 4u);
        async_ld_b128(ldsB + 4096u + (unsigned)tid * 16u,
                      logits + base + 1024u + (size_t)tid * 4u);
        async_ld_b64(ldsB + 8192u + (unsigned)tid * 8u,
                     logits + base + 2048u + (size_t)tid * 2u);
    };

    if (blockIdx.x < (unsigned)fullTiles) {
        issue_tile(tile[0], blockIdx.x);                    // prologue: prime buf0
        int k = 0;
        for (int t = blockIdx.x; t < fullTiles; t += g, ++k) {
            const float* cur = tile[k & 1];
            const float* nxt = tile[(k + 1) & 1];
            const bool hasNext = (t + g) < fullTiles;
            if (hasNext) {
                issue_tile(nxt, t + g);   // safe: nxt fully consumed at iter k-1 (+barrier)
            }
            int label = labels[t * TILE_ROWS + tid];        // overlaps with async
            if (hasNext) wait_async3();   // in-order completion: tile t's 3 ops are done
            else         wait_async0();
            __syncthreads();

            bool good;
            int bin = classify_row(&cur[tid * C], label, good);

            // wave32 ballot reduction: one conflict-free LDS add per bin per wave
            unsigned myCnt = 0, myCor = 0;
#pragma unroll
            for (int b = 0; b < NBINS; ++b) {
                unsigned mc = ballot32(bin == b);
                unsigned mg = ballot32(good && (bin == b));
                if (lane == b) {
                    myCnt = (unsigned)__builtin_popcount(mc);
                    myCor = (unsigned)__builtin_popcount(mg);
                }
            }
            if (lane < NBINS) {
                if (myCnt) atomicAdd(&s_cnt[lane], myCnt);
                if (myCor) atomicAdd(&s_cor[lane], myCor);
            }
            __syncthreads();  // all waves done reading `cur` before it is refilled
        }
    }

    // Tail rows (N % 256), handled by block 0 with direct guarded loads.
    const int tailStart = fullTiles * TILE_ROWS;
    if (blockIdx.x == 0 && tailStart < n) {
        int r = tailStart + tid;
        if (r < n) {
            float row[C];
            const float* rp = logits + (size_t)r * C;
#pragma unroll
            for (int j = 0; j < C; ++j) row[j] = rp[j];
            bool good;
            int bin = classify_row(row, labels[r], good);
            if (bin >= 0) {
                atomicAdd(&s_cnt[bin], 1u);
                if (good) atomicAdd(&s_cor[bin], 1u);
            }
        }
    }

    __syncthreads();
    // Publish this block's partials: cols [0,10)=counts, [16,26)=correct, rest 0.
    if (tid < 32) {
        unsigned v = 0;
        if (tid < NBINS) v = s_cnt[tid];
        else if (tid >= 16 && tid < 16 + NBINS) v = s_cor[tid - 16];
        partials[(size_t)blockIdx.x * 32u + (unsigned)tid] = v;
    }
}

__global__ void __launch_bounds__(256) calib_finalize(const unsigned* __restrict__ partials,
                                                      float* __restrict__ out,
                                                      int nblk) {
    __shared__ unsigned sums[32];
    const int t = threadIdx.x;
    if (t < 32) sums[t] = 0u;
    __syncthreads();
    const unsigned c = (unsigned)(t & 31);
    unsigned s = 0;
    for (int r = (t >> 5); r < nblk; r += (256 / 32))
        s += partials[(size_t)r * 32u + c];
    atomicAdd(&sums[c], s);
    __syncthreads();
    if (t < NBINS) {
        unsigned cnt = sums[t];
        unsigned cor = sums[16 + t];
        out[t] = (cnt > 0u) ? ((float)cor / (float)cnt) : 0.0f;
    }
}

extern "C" void kernel_launch(void* const* d_in, const int* in_sizes, int n_in,
                              void* d_out, int out_size, void* d_ws, size_t ws_size,
                              hipStream_t stream) {
    const float* logits = (const float*)d_in[0];
    const int*   labels = (const int*)d_in[1];   // harness passes integers as int32
    const int n = in_sizes[1];                   // N rows = label count

    unsigned* partials = (unsigned*)d_ws;
    int nblk = NBLK_MAX;
    const size_t need = (size_t)NBLK_MAX * 32u * sizeof(unsigned);  // 128 KB
    if (ws_size < need) {
        nblk = (int)(ws_size / (32u * sizeof(unsigned)));
        if (nblk < 1) nblk = 1;
    }

    calib_main<<<nblk, TPB, 0, stream>>>(logits, labels, partials, n);
    calib_finalize<<<1, 256, 0, stream>>>(partials, (float*)d_out, nblk);
}